// HHGNN_30167850287705
// MI455X (gfx1250) — compile-verified
//
#include <hip/hip_runtime.h>
#include <math.h>

#define DRUG_NUM   38
#define CLINE_NUM  32
#define NUM_NODE   70
#define FEA        128
#define NUM_CLS    131072
#define M_ENTRIES  524288
#define BATCH      8192
#define NSEG       140          // 70 nodes * 2 groups

typedef float v2f __attribute__((ext_vector_type(2)));
typedef float v8f __attribute__((ext_vector_type(8)));

// -------- workspace layout (float offsets) --------
#define WS_SUMS   0             // 140*70      = 9800
#define WS_CNTS   9800          // 140
#define WS_EG     9984          // 140*128     = 17920   edge_group (mean)
#define WS_NODEM  27904         // 70*128      = 8960
#define WS_EFEAT  36864         // 140*128     = 17920
#define WS_EDGEM  54784         // 140*128     = 17920
#define WS_NODER  72704         // 70*128      = 8960
#define WS_W1T    81920         // 384*512     = 196608
#define WS_W2T    278528        // 512*256     = 131072
#define WS_H1     409600        // 8192*512    = 4194304

// ---------------------------------------------------------------------------
// 0) zero the segment accumulators
// ---------------------------------------------------------------------------
__global__ void k_init(float* __restrict__ ws) {
    int gid = blockIdx.x * blockDim.x + threadIdx.x;
    if (gid < NSEG * NUM_NODE + NSEG) ws[WS_SUMS + gid] = 0.0f;
}

// ---------------------------------------------------------------------------
// 1) transpose decoder weights:  w1t[k*512+n] = lin1_w[n*384+k], w2t likewise
// ---------------------------------------------------------------------------
__global__ void k_transpose(const float* __restrict__ w1,
                            const float* __restrict__ w2,
                            float* __restrict__ ws) {
    int gid = blockIdx.x * blockDim.x + threadIdx.x;
    if (gid < 384 * 512) {
        int n = gid & 511, k = gid >> 9;
        ws[WS_W1T + gid] = w1[n * 384 + k];
    }
    int g2 = gid - 384 * 512;
    if (g2 >= 0 && g2 < 512 * 256) {
        int n = g2 & 255, k = g2 >> 8;
        ws[WS_W2T + g2] = w2[n * 512 + k];
    }
}

// ---------------------------------------------------------------------------
// 2) segment-sum of DTH rows:  sums[seg][0:70] += DTH_row(he_edge), cnt[seg]++
//    Wave-uniform entries: all 32 lanes read one 280B row coalesced and
//    ds_add_f32 into LDS accumulators; block flushes with global fp atomics.
// ---------------------------------------------------------------------------
#define SEG_BLOCKS 256
__global__ __launch_bounds__(256)
void k_segsum(const float* __restrict__ dth_cls,
              const float* __restrict__ dth_dc,
              const int* __restrict__ he_node,
              const int* __restrict__ he_edge,
              float* __restrict__ ws) {
    __shared__ float lsum[NSEG * NUM_NODE];
    __shared__ float lcnt[NSEG];
    for (int i = threadIdx.x; i < NSEG * NUM_NODE; i += 256) lsum[i] = 0.0f;
    for (int i = threadIdx.x; i < NSEG; i += 256) lcnt[i] = 0.0f;
    __syncthreads();

    const int lane   = threadIdx.x & 31;
    const int gwave  = (blockIdx.x * 256 + threadIdx.x) >> 5;
    const int nwaves = SEG_BLOCKS * 8;
    const int per    = M_ENTRIES / nwaves;          // 256
    const int base   = gwave * per;

    for (int i = 0; i < per; ++i) {
        int m  = base + i;
        int hn = he_node[m];
        int he = he_edge[m];
        int grp = (he >= NUM_CLS) ? 1 : 0;
        int seg = hn * 2 + grp;
        const float* row = grp ? (dth_dc + (he - NUM_CLS) * NUM_NODE)
                               : (dth_cls + he * NUM_NODE);
        float* dst = lsum + seg * NUM_NODE;
        atomicAdd(&dst[lane],      row[lane]);
        atomicAdd(&dst[lane + 32], row[lane + 32]);
        if (lane < NUM_NODE - 64) atomicAdd(&dst[lane + 64], row[lane + 64]);
        if (lane == 0) atomicAdd(&lcnt[seg], 1.0f);
    }
    __syncthreads();

    for (int i = threadIdx.x; i < NSEG * NUM_NODE; i += 256)
        unsafeAtomicAdd(&ws[WS_SUMS + i], lsum[i]);
    for (int i = threadIdx.x; i < NSEG; i += 256)
        unsafeAtomicAdd(&ws[WS_CNTS + i], lcnt[i]);
}

// ---------------------------------------------------------------------------
// 3) edge_group = (sums @ merge) / cnt   and   node_m = merge @ node_proj[t]
// ---------------------------------------------------------------------------
__global__ void k_eg_nodem(const float* __restrict__ dEmb,
                           const float* __restrict__ cEmb,
                           const float* __restrict__ node_proj,
                           float* __restrict__ ws) {
    int gid = blockIdx.x * blockDim.x + threadIdx.x;
    if (gid < NSEG * FEA) {
        int r = gid >> 7, c = gid & 127;
        const float* s = ws + WS_SUMS + r * NUM_NODE;
        float acc = 0.0f;
        for (int j = 0; j < DRUG_NUM; ++j)  acc += s[j] * dEmb[j * FEA + c];
        for (int j = 0; j < CLINE_NUM; ++j) acc += s[DRUG_NUM + j] * cEmb[j * FEA + c];
        float cnt = ws[WS_CNTS + r];
        cnt = cnt > 1.0f ? cnt : 1.0f;
        ws[WS_EG + gid] = acc / cnt;
    } else if (gid < NSEG * FEA + NUM_NODE * FEA) {
        int o = gid - NSEG * FEA;
        int n = o >> 7, c = o & 127;
        int t = (n < DRUG_NUM) ? 0 : 1;
        const float* row = (n < DRUG_NUM) ? (dEmb + n * FEA)
                                          : (cEmb + (n - DRUG_NUM) * FEA);
        const float* P = node_proj + t * FEA * FEA;
        float acc = 0.0f;
        for (int j = 0; j < FEA; ++j) acc += row[j] * P[j * FEA + c];
        ws[WS_NODEM + o] = acc;
    }
}

// ---------------------------------------------------------------------------
// 4) edge_feat = relu(edge_group @ fc_w^T + fc_b)
//    edge_m    = edge_group @ edge_proj[p]    (p = 2*(n>=38) + g)
// ---------------------------------------------------------------------------
__global__ void k_efeat_em(const float* __restrict__ fc_w,
                           const float* __restrict__ fc_b,
                           const float* __restrict__ edge_proj,
                           float* __restrict__ ws) {
    int gid = blockIdx.x * blockDim.x + threadIdx.x;
    int half = NSEG * FEA;
    if (gid < half) {
        int r = gid >> 7, c = gid & 127;
        const float* e = ws + WS_EG + r * FEA;
        const float* w = fc_w + c * FEA;
        float acc = fc_b[c];
        for (int j = 0; j < FEA; ++j) acc += e[j] * w[j];
        ws[WS_EFEAT + gid] = acc > 0.0f ? acc : 0.0f;
    } else if (gid < 2 * half) {
        int o = gid - half;
        int r = o >> 7, c = o & 127;
        int n = r >> 1, g = r & 1;
        int p = ((n < DRUG_NUM) ? 0 : 2) + g;
        const float* e = ws + WS_EG + r * FEA;
        const float* P = edge_proj + p * FEA * FEA;
        float acc = 0.0f;
        for (int j = 0; j < FEA; ++j) acc += e[j] * P[j * FEA + c];
        ws[WS_EDGEM + o] = acc;
    }
}

// ---------------------------------------------------------------------------
// 5) attention scores, softmax over HEAD axis, node_rep (single block)
// ---------------------------------------------------------------------------
__global__ __launch_bounds__(256)
void k_attn(float* __restrict__ ws) {
    __shared__ float sc[NUM_NODE][4][2];
    __shared__ float wgt[NUM_NODE][4][2];
    const float* node_m = ws + WS_NODEM;
    const float* edge_m = ws + WS_EDGEM;
    const float* efeat  = ws + WS_EFEAT;

    for (int s = threadIdx.x; s < NUM_NODE * 8; s += 256) {
        int n = s >> 3, h = (s >> 1) & 3, e = s & 1;
        const float* q = node_m + n * FEA + h * 32;
        const float* k = edge_m + (n * 2 + e) * FEA + h * 32;
        float acc = 0.0f;
        for (int d = 0; d < 32; ++d) acc += q[d] * k[d];
        sc[n][h][e] = acc * 0.17677669529663687f;   // 1/sqrt(32)
    }
    __syncthreads();
    for (int p = threadIdx.x; p < NUM_NODE * 2; p += 256) {
        int n = p >> 1, e = p & 1;
        float mx = sc[n][0][e];
        for (int h = 1; h < 4; ++h) mx = fmaxf(mx, sc[n][h][e]);
        float den = 0.0f;
        float ex[4];
        for (int h = 0; h < 4; ++h) { ex[h] = expf(sc[n][h][e] - mx); den += ex[h]; }
        for (int h = 0; h < 4; ++h) wgt[n][h][e] = ex[h] / den;
    }
    __syncthreads();
    for (int o = threadIdx.x; o < NUM_NODE * FEA; o += 256) {
        int n = o >> 7, c = o & 127, h = c >> 5;
        float x0 = wgt[n][h][0] * efeat[(n * 2 + 0) * FEA + c];
        float x1 = wgt[n][h][1] * efeat[(n * 2 + 1) * FEA + c];
        x0 = x0 > 0.0f ? x0 : 0.0f;
        x1 = x1 > 0.0f ? x1 : 0.0f;
        ws[WS_NODER + o] = x0 + x1;
    }
}

// ---------------------------------------------------------------------------
// 6) decoder GEMM1 (WMMA f32 16x16x4): gather + BN fused into A staging,
//    leaky-relu(0.5) fused into C store.  h1[8192,512]
// ---------------------------------------------------------------------------
__global__ __launch_bounds__(256)
void k_gemm1(const int* __restrict__ idx,
             const float* __restrict__ bn_g, const float* __restrict__ bn_b,
             const float* __restrict__ bn_m, const float* __restrict__ bn_v,
             const float* __restrict__ lin1_b,
             float* __restrict__ ws) {
    __shared__ float As[16][385];
    const float* node_rep = ws + WS_NODER;
    const float* w1t      = ws + WS_W1T;
    float*       h1       = ws + WS_H1;
    const int m0 = blockIdx.x * 16;

    for (int i = threadIdx.x; i < 16 * 384; i += 256) {
        int r = i / 384, k = i - r * 384;
        int j = k >> 7, col = k & 127;
        int node = idx[(m0 + r) * 3 + j];
        float v = node_rep[node * FEA + col];
        v = (v - bn_m[k]) * rsqrtf(bn_v[k] + 1e-5f) * bn_g[k] + bn_b[k];
        As[r][k] = v;
    }
    __syncthreads();

    const int wave = threadIdx.x >> 5;
    const int lane = threadIdx.x & 31;
    const int r    = lane & 15;
    const int kh   = (lane >> 4) << 1;           // 0 or 2
    for (int t = 0; t < 4; ++t) {
        const int n0 = (wave * 4 + t) * 16;
        v8f c = {};
        for (int k0 = 0; k0 < 384; k0 += 4) {
            v2f a, b;
            a.x = As[r][k0 + kh];
            a.y = As[r][k0 + kh + 1];
            b.x = w1t[(k0 + kh) * 512 + n0 + r];
            b.y = w1t[(k0 + kh + 1) * 512 + n0 + r];
            c = __builtin_amdgcn_wmma_f32_16x16x4_f32(
                    false, a, false, b, (short)0, c, false, false);
        }
        const float bias = lin1_b[n0 + r];
        const int mbase = m0 + ((lane >> 4) << 3);
        #pragma unroll
        for (int v = 0; v < 8; ++v) {
            float x = c[v] + bias;
            x = x > 0.0f ? x : 0.5f * x;
            h1[(mbase + v) * 512 + n0 + r] = x;
        }
    }
}

// ---------------------------------------------------------------------------
// 7) decoder GEMM2 (WMMA f32 16x16x4): h2 = leaky(h1 @ w2t + b2) -> d_out+8192
// ---------------------------------------------------------------------------
__global__ __launch_bounds__(256)
void k_gemm2(const float* __restrict__ lin2_b,
             float* __restrict__ ws, float* __restrict__ out) {
    __shared__ float As[16][513];
    const float* h1  = ws + WS_H1;
    const float* w2t = ws + WS_W2T;
    float*       h2  = out + BATCH;
    const int m0 = blockIdx.x * 16;

    for (int i = threadIdx.x; i < 16 * 512; i += 256) {
        int r = i >> 9, k = i & 511;
        As[r][k] = h1[(m0 + r) * 512 + k];
    }
    __syncthreads();

    const int wave = threadIdx.x >> 5;
    const int lane = threadIdx.x & 31;
    const int r    = lane & 15;
    const int kh   = (lane >> 4) << 1;
    for (int t = 0; t < 2; ++t) {
        const int n0 = (wave * 2 + t) * 16;
        v8f c = {};
        for (int k0 = 0; k0 < 512; k0 += 4) {
            v2f a, b;
            a.x = As[r][k0 + kh];
            a.y = As[r][k0 + kh + 1];
            b.x = w2t[(k0 + kh) * 256 + n0 + r];
            b.y = w2t[(k0 + kh + 1) * 256 + n0 + r];
            c = __builtin_amdgcn_wmma_f32_16x16x4_f32(
                    false, a, false, b, (short)0, c, false, false);
        }
        const float bias = lin2_b[n0 + r];
        const int mbase = m0 + ((lane >> 4) << 3);
        #pragma unroll
        for (int v = 0; v < 8; ++v) {
            float x = c[v] + bias;
            x = x > 0.0f ? x : 0.5f * x;
            h2[(mbase + v) * 256 + n0 + r] = x;
        }
    }
}

// ---------------------------------------------------------------------------
// 8) final matvec + sigmoid: out[b] = sigmoid(h2[b] . lin3_w + lin3_b)
//    one wave per row, shuffle reduction
// ---------------------------------------------------------------------------
__global__ __launch_bounds__(256)
void k_final(const float* __restrict__ lin3_w, const float* __restrict__ lin3_b,
             float* __restrict__ out) {
    const float* h2 = out + BATCH;
    const int lane = threadIdx.x & 31;
    const int b = blockIdx.x * 8 + (threadIdx.x >> 5);
    float acc = 0.0f;
    #pragma unroll
    for (int i = 0; i < 8; ++i) {
        int c = lane + i * 32;
        acc += h2[b * 256 + c] * lin3_w[c];
    }
    #pragma unroll
    for (int off = 16; off >= 1; off >>= 1)
        acc += __shfl_xor(acc, off, 32);
    if (lane == 0) {
        float ret = acc + lin3_b[0];
        out[b] = 1.0f / (1.0f + expf(-ret));
    }
}

// ---------------------------------------------------------------------------
extern "C" void kernel_launch(void* const* d_in, const int* in_sizes, int n_in,
                              void* d_out, int out_size, void* d_ws, size_t ws_size,
                              hipStream_t stream) {
    const float* DTH_cls  = (const float*)d_in[2];
    const float* DTH_dc   = (const float*)d_in[3];
    const int*   he_node  = (const int*)d_in[4];
    const int*   he_edge  = (const int*)d_in[5];
    const int*   index    = (const int*)d_in[7];
    const float* dEmb     = (const float*)d_in[8];
    const float* cEmb     = (const float*)d_in[9];
    const float* node_prj = (const float*)d_in[10];
    const float* edge_prj = (const float*)d_in[11];
    const float* fc_w     = (const float*)d_in[12];
    const float* fc_b     = (const float*)d_in[13];
    const float* bn_g     = (const float*)d_in[14];
    const float* bn_b     = (const float*)d_in[15];
    const float* bn_m     = (const float*)d_in[16];
    const float* bn_v     = (const float*)d_in[17];
    const float* lin1_w   = (const float*)d_in[18];
    const float* lin1_b   = (const float*)d_in[19];
    const float* lin2_w   = (const float*)d_in[20];
    const float* lin2_b   = (const float*)d_in[21];
    const float* lin3_w   = (const float*)d_in[22];
    const float* lin3_b   = (const float*)d_in[23];
    float* ws  = (float*)d_ws;
    float* out = (float*)d_out;

    k_init<<<(NSEG * NUM_NODE + NSEG + 255) / 256, 256, 0, stream>>>(ws);
    k_transpose<<<(384 * 512 + 512 * 256 + 255) / 256, 256, 0, stream>>>(lin1_w, lin2_w, ws);
    k_segsum<<<SEG_BLOCKS, 256, 0, stream>>>(DTH_cls, DTH_dc, he_node, he_edge, ws);
    k_eg_nodem<<<(NSEG * FEA + NUM_NODE * FEA + 255) / 256, 256, 0, stream>>>(
        dEmb, cEmb, node_prj, ws);
    k_efeat_em<<<(2 * NSEG * FEA + 255) / 256, 256, 0, stream>>>(fc_w, fc_b, edge_prj, ws);
    k_attn<<<1, 256, 0, stream>>>(ws);
    k_gemm1<<<BATCH / 16, 256, 0, stream>>>(index, bn_g, bn_b, bn_m, bn_v, lin1_b, ws);
    k_gemm2<<<BATCH / 16, 256, 0, stream>>>(lin2_b, ws, out);
    k_final<<<BATCH / 8, 256, 0, stream>>>(lin3_w, lin3_b, out);
}